// GCN_32899449487582
// MI455X (gfx1250) — compile-verified
//
#include <hip/hip_runtime.h>
#include <hip/hip_bf16.h>

typedef float v2f __attribute__((ext_vector_type(2)));
typedef float v8f __attribute__((ext_vector_type(8)));

#define HF 128          // h_feats / embedding size
#define LPITCH 132      // LDS row pitch (bank-conflict free: 132 % 64 = 4)
#define EPSBN 1e-5f

__device__ __forceinline__ void atomAddF(float* p, float v) {
    __hip_atomic_fetch_add(p, v, __ATOMIC_RELAXED, __HIP_MEMORY_SCOPE_AGENT);
}

// ---------------- utility ----------------
__global__ void zerof_kernel(float* __restrict__ p, int n) {
    int i = blockIdx.x * blockDim.x + threadIdx.x;
    if (i < n) p[i] = 0.0f;
}

// deg[dst[e]] += 1
__global__ void deg_kernel(const int* __restrict__ dst, float* __restrict__ deg, int e) {
    int i = blockIdx.x * blockDim.x + threadIdx.x;
    if (i < e) atomAddF(&deg[dst[i]], 1.0f);
}

// cnt[graph_ids[n]] += 1
__global__ void cnt_kernel(const int* __restrict__ gid, float* __restrict__ cnt, int n) {
    int i = blockIdx.x * blockDim.x + threadIdx.x;
    if (i < n) atomAddF(&cnt[gid[i]], 1.0f);
}

// h[n,:] = emb[in_feat[n],:]   (32 lanes per node, float4 each)
__global__ void gather_emb_kernel(const int* __restrict__ feat, const float* __restrict__ emb,
                                  float* __restrict__ h, int n) {
    int idx = blockIdx.x * blockDim.x + threadIdx.x;
    if (idx >= n * 32) return;
    int r  = idx >> 5;
    int c4 = (idx & 31) << 2;
    const float4 v = *(const float4*)(emb + (long long)feat[r] * HF + c4);
    *(float4*)(h + (long long)r * HF + c4) = v;
}

// agg[dst[e],:] += h[src[e],:]   (32 lanes per edge, float4 load, 4 f32 atomics)
__global__ void edge_scatter_kernel(const int* __restrict__ src, const int* __restrict__ dst,
                                    const float* __restrict__ h, float* __restrict__ agg, int e) {
    int idx = blockIdx.x * blockDim.x + threadIdx.x;
    if (idx >= e * 32) return;
    int ei = idx >> 5;
    int c4 = (idx & 31) << 2;
    int s = src[ei], d = dst[ei];
    const float4 v = *(const float4*)(h + (long long)s * HF + c4);
    float* ap = agg + (long long)d * HF + c4;
    atomAddF(ap + 0, v.x);
    atomAddF(ap + 1, v.y);
    atomAddF(ap + 2, v.z);
    atomAddF(ap + 3, v.w);
}

// rst = PReLU(h @ Wself + (agg/deg) @ Wneigh + b); accumulate per-channel sum/sumsq.
// Block = 256 threads = 8 wave32; block owns 16 rows; wave w owns cols [16w,16w+16).
__global__ void __launch_bounds__(256)
sage_gemm_prelu_kernel(const float* __restrict__ h, const float* __restrict__ agg,
                       const float* __restrict__ deg,
                       const float* __restrict__ Wself, const float* __restrict__ Wneigh,
                       const float* __restrict__ bias, const float* __restrict__ pw,
                       float* __restrict__ rst, float* __restrict__ stats, int n) {
    __shared__ float shH[16 * LPITCH];
    __shared__ float shA[16 * LPITCH];
    __shared__ float shSum[256];
    __shared__ float shSq[256];

    const int tid  = threadIdx.x;
    const int row0 = blockIdx.x * 16;

    // Stage 16 rows of h and normalized agg into LDS (zero-pad past N).
    for (int i = tid; i < 16 * HF; i += 256) {
        int r = i >> 7, c = i & (HF - 1);
        int gr = row0 + r;
        float hv = 0.0f, av = 0.0f;
        if (gr < n) {
            hv = h[(long long)gr * HF + c];
            float d = deg[gr];
            d = d < 1.0f ? 1.0f : d;
            av = agg[(long long)gr * HF + c] / d;
        }
        shH[r * LPITCH + c] = hv;
        shA[r * LPITCH + c] = av;
    }
    __syncthreads();

    const int wave = tid >> 5;
    const int lane = tid & 31;
    const int half = lane >> 4;      // K sub-split for A/B, M sub-split for C/D
    const int lr   = lane & 15;      // A row / B,C,D column-in-tile
    const int col  = (wave << 4) + lr;

    v8f acc = {};
    // h @ W_self : K = 128 in 32 steps of 4
    for (int kk = 0; kk < 32; ++kk) {
        int k0 = 4 * kk + 2 * half;
        v2f a = { shH[lr * LPITCH + k0], shH[lr * LPITCH + k0 + 1] };
        v2f b = { Wself[k0 * HF + col], Wself[(k0 + 1) * HF + col] };
        acc = __builtin_amdgcn_wmma_f32_16x16x4_f32(false, a, false, b,
                                                    (short)0, acc, false, false);
    }
    // (agg/deg) @ W_neigh accumulated into same C
    for (int kk = 0; kk < 32; ++kk) {
        int k0 = 4 * kk + 2 * half;
        v2f a = { shA[lr * LPITCH + k0], shA[lr * LPITCH + k0 + 1] };
        v2f b = { Wneigh[k0 * HF + col], Wneigh[(k0 + 1) * HF + col] };
        acc = __builtin_amdgcn_wmma_f32_16x16x4_f32(false, a, false, b,
                                                    (short)0, acc, false, false);
    }

    const float bv = bias[col];
    const float pv = pw[col];
    float psum = 0.0f, psq = 0.0f;
    for (int i = 0; i < 8; ++i) {
        int gr = row0 + half * 8 + i;          // C/D layout: M = i + 8*half
        if (gr < n) {
            float v = acc[i] + bv;
            v = v > 0.0f ? v : pv * v;          // PReLU (per-channel)
            rst[(long long)gr * HF + col] = v;
            psum += v;
            psq  += v * v;
        }
    }
    // Block reduction of per-channel partial stats, then one atomic per channel.
    shSum[tid] = psum;
    shSq[tid]  = psq;
    __syncthreads();
    if (half == 0) {
        float s = shSum[tid] + shSum[tid + 16];
        float q = shSq[tid]  + shSq[tid + 16];
        atomAddF(&stats[col], s);
        atomAddF(&stats[HF + col], q);
    }
}

// BatchNorm (training-mode batch stats); write next-layer h and the concat slice.
__global__ void bn_apply_kernel(const float* __restrict__ rst, const float* __restrict__ stats,
                                const float* __restrict__ gamma, const float* __restrict__ beta,
                                float* __restrict__ hout, float* __restrict__ outcat,
                                int layer_off, int n, float invN) {
    int idx = blockIdx.x * blockDim.x + threadIdx.x;
    if (idx >= n * HF) return;
    int c = idx & (HF - 1);
    int r = idx >> 7;
    float mu  = stats[c] * invN;
    float var = stats[HF + c] * invN - mu * mu;
    float v = (rst[idx] - mu) * rsqrtf(var + EPSBN) * gamma[c] + beta[c];
    hout[idx] = v;
    outcat[(long long)r * (4 * HF) + layer_off + c] = v;
}

// gout[graph_ids[n],:] += hcat[n,:]
__global__ void pool_accum_kernel(const float* __restrict__ outcat, const int* __restrict__ gid,
                                  float* __restrict__ gout, int n) {
    int idx = blockIdx.x * blockDim.x + threadIdx.x;
    if (idx >= n * HF) return;                    // 128 lanes/node, 4 chans each (512 total)
    int r  = idx >> 7;
    int c4 = (idx & (HF - 1)) << 2;
    int g  = gid[r];
    const float4 v = *(const float4*)(outcat + (long long)r * (4 * HF) + c4);
    float* gp = gout + (long long)g * (4 * HF) + c4;
    atomAddF(gp + 0, v.x);
    atomAddF(gp + 1, v.y);
    atomAddF(gp + 2, v.z);
    atomAddF(gp + 3, v.w);
}

__global__ void pool_final_kernel(float* __restrict__ gout, const float* __restrict__ cnt, int total) {
    int idx = blockIdx.x * blockDim.x + threadIdx.x;
    if (idx >= total) return;
    int g = idx >> 9;                             // 512 channels per graph
    float c = cnt[g];
    c = c < 1.0f ? 1.0f : c;
    gout[idx] /= c;
}

extern "C" void kernel_launch(void* const* d_in, const int* in_sizes, int n_in,
                              void* d_out, int out_size, void* d_ws, size_t ws_size,
                              hipStream_t stream) {
    const int*   in_feat = (const int*)d_in[0];
    const int*   src     = (const int*)d_in[1];
    const int*   dst     = (const int*)d_in[2];
    const int*   gids    = (const int*)d_in[3];
    const float* emb     = (const float*)d_in[4];
    const float* W_self  = (const float*)d_in[5];
    const float* W_neigh = (const float*)d_in[6];
    const float* bias    = (const float*)d_in[7];
    const float* gamma   = (const float*)d_in[8];
    const float* beta    = (const float*)d_in[9];
    const float* prelu   = (const float*)d_in[10];
    float* gout = (float*)d_out;

    const int N = in_sizes[0];
    const int E = in_sizes[1];
    const int L = in_sizes[5] / (HF * HF);
    const int G = out_size / (4 * HF);

    // Workspace carve (256B aligned)
    auto align_up = [](size_t x) { return (x + 255) & ~(size_t)255; };
    char* base = (char*)d_ws;
    size_t off = 0;
    float* h      = (float*)(base + off); off = align_up(off + (size_t)N * HF * 4);
    float* agg    = (float*)(base + off); off = align_up(off + (size_t)N * HF * 4);
    float* rst    = (float*)(base + off); off = align_up(off + (size_t)N * HF * 4);
    float* outcat = (float*)(base + off); off = align_up(off + (size_t)N * 4 * HF * 4);
    float* deg    = (float*)(base + off); off = align_up(off + (size_t)N * 4);
    float* cnt    = (float*)(base + off); off = align_up(off + (size_t)G * 4);
    float* stats  = (float*)(base + off); off = align_up(off + 2 * HF * 4);
    (void)ws_size; (void)n_in;

    const int TB = 256;
    auto blk = [](long long t, int b) { return (int)((t + b - 1) / b); };

    // Degrees, graph counts, embedding gather
    zerof_kernel<<<blk(N, TB), TB, 0, stream>>>(deg, N);
    zerof_kernel<<<blk(G, TB), TB, 0, stream>>>(cnt, G);
    deg_kernel<<<blk(E, TB), TB, 0, stream>>>(dst, deg, E);
    cnt_kernel<<<blk(N, TB), TB, 0, stream>>>(gids, cnt, N);
    gather_emb_kernel<<<blk((long long)N * 32, TB), TB, 0, stream>>>(in_feat, emb, h, N);

    const float invN = 1.0f / (float)N;
    for (int l = 0; l < L; ++l) {
        zerof_kernel<<<blk((long long)N * HF, TB), TB, 0, stream>>>(agg, N * HF);
        zerof_kernel<<<1, 256, 0, stream>>>(stats, 2 * HF);
        edge_scatter_kernel<<<blk((long long)E * 32, TB), TB, 0, stream>>>(src, dst, h, agg, E);
        sage_gemm_prelu_kernel<<<blk(N, 16), 256, 0, stream>>>(
            h, agg, deg,
            W_self + (size_t)l * HF * HF, W_neigh + (size_t)l * HF * HF,
            bias + (size_t)l * HF, prelu + (size_t)l * HF,
            rst, stats, N);
        bn_apply_kernel<<<blk((long long)N * HF, TB), TB, 0, stream>>>(
            rst, stats, gamma + (size_t)l * HF, beta + (size_t)l * HF,
            h, outcat, l * HF, N, invN);
    }

    // Graph mean pooling into d_out
    zerof_kernel<<<blk((long long)G * 4 * HF, TB), TB, 0, stream>>>(gout, G * 4 * HF);
    pool_accum_kernel<<<blk((long long)N * HF, TB), TB, 0, stream>>>(outcat, gids, gout, N);
    pool_final_kernel<<<blk((long long)G * 4 * HF, TB), TB, 0, stream>>>(gout, cnt, G * 4 * HF);
}